// HashEmbedding_69148973466026
// MI455X (gfx1250) — compile-verified
//
#include <hip/hip_runtime.h>
#include <cstdint>
#include <cstddef>

// ---------------------------------------------------------------------------
// InstantNGP hash-grid embedding, MI455X (gfx1250, wave32).
// Thread = (point, level). Block = 256 threads = 16 points x 16 levels.
// Coordinates staged global->LDS via CDNA5 async copy (ASYNCcnt), gathers are
// L2-resident global_load_b64, output written with non-temporal b64 stores so
// the 134MB output stream does not evict the 64MB of tables from the 192MB L2.
// ---------------------------------------------------------------------------

#define NGP_N_LEVELS   16
#define NGP_LOG2_T     19
#define NGP_TABLE_SIZE (1u << NGP_LOG2_T)
#define NGP_MASK       (NGP_TABLE_SIZE - 1u)
#define NGP_P1         2654435761u
#define NGP_P2         805459861u
#define NGP_PTS_PER_BLK 16
#define NGP_THREADS     256

// floor(16 * (512/16)^(i/15)) = floor(16 * 2^(i/3))
__constant__ int c_ngp_res[NGP_N_LEVELS] = {
    16, 20, 25, 32, 40, 50, 64, 80, 101, 128, 161, 203, 256, 322, 406, 512};

typedef float v2f __attribute__((ext_vector_type(2)));

__device__ __forceinline__ unsigned ngp_lds_off(const void* p) {
    // generic -> LDS(as3) pointer; numeric value of an as3 pointer is the
    // byte offset within this workgroup's LDS allocation (what the async
    // load's VDST operand expects).
    return (unsigned)(uintptr_t)(__attribute__((address_space(3))) const void*)p;
}

__global__ __launch_bounds__(NGP_THREADS) void HashEmbedding_ngp_kernel(
    const float* __restrict__ x,      // (B, 3)
    const float* __restrict__ emb,    // (16, 2^19, 2)
    float* __restrict__ out,          // (B, 32)
    int npts)
{
    __shared__ __align__(16) float sc[NGP_PTS_PER_BLK * 4];  // xyz + pad per point

    const int tid      = (int)threadIdx.x;
    const int ptLocal  = tid >> 4;     // 0..15 point within block
    const int lvl      = tid & 15;     // 0..15 level
    const long long basePt = (long long)blockIdx.x * NGP_PTS_PER_BLK;
    const int nvalid   = min(NGP_PTS_PER_BLK, npts - (int)basePt);

    // ---- stage 16 points x 3 floats into LDS via CDNA5 async copy ----------
    if (tid < NGP_PTS_PER_BLK * 3 && tid < nvalid * 3) {
        const int pt = tid / 3;
        const int cc = tid - pt * 3;
        const unsigned loff =
            ngp_lds_off(sc) + (unsigned)(pt * 16 + cc * 4);
        const unsigned long long ga =
            (unsigned long long)(const void*)(x + basePt * 3 + tid);
        asm volatile("global_load_async_to_lds_b32 %0, %1, off"
                     :: "v"(loff), "v"(ga) : "memory");
    }
    asm volatile("s_wait_asynccnt 0x0" ::: "memory");
    __syncthreads();

    if (ptLocal >= nvalid) return;

    // one ds_load_b128 per thread (16-lane broadcast within a level group)
    const float4 c4 = *reinterpret_cast<const float4*>(&sc[ptLocal * 4]);

    // ---- per-level interpolation -------------------------------------------
    const float fres = (float)c_ngp_res[lvl];
    const float gs   = 2.0f / fres;                 // (GRID_MAX-GRID_MIN)/res
    const float rx   = (c4.x + 1.0f) / gs;          // (x - GRID_MIN)/grid_size
    const float ry   = (c4.y + 1.0f) / gs;
    const float rz   = (c4.z + 1.0f) / gs;

    const float hi = fres - 1.0f;
    const float fx = fminf(fmaxf(floorf(rx), 0.0f), hi);
    const float fy = fminf(fmaxf(floorf(ry), 0.0f), hi);
    const float fz = fminf(fmaxf(floorf(rz), 0.0f), hi);

    const float wx = rx - fx, wy = ry - fy, wz = rz - fz;

    const uint32_t ix = (uint32_t)fx, iy = (uint32_t)fy, iz = (uint32_t)fz;

    // (i+1)*P == i*P + P mod 2^32
    const uint32_t ax0 = ix;            const uint32_t ax1 = ix + 1u;
    const uint32_t ay0 = iy * NGP_P1;   const uint32_t ay1 = ay0 + NGP_P1;
    const uint32_t az0 = iz * NGP_P2;   const uint32_t az1 = az0 + NGP_P2;

    const uint32_t lbase = (uint32_t)lvl << NGP_LOG2_T;
    const v2f* __restrict__ tab = reinterpret_cast<const v2f*>(emb);

    const uint32_t h0 = lbase + ((ax0 ^ ay0 ^ az0) & NGP_MASK);
    const uint32_t h1 = lbase + ((ax1 ^ ay0 ^ az0) & NGP_MASK);
    const uint32_t h2 = lbase + ((ax0 ^ ay1 ^ az0) & NGP_MASK);
    const uint32_t h3 = lbase + ((ax1 ^ ay1 ^ az0) & NGP_MASK);
    const uint32_t h4 = lbase + ((ax0 ^ ay0 ^ az1) & NGP_MASK);
    const uint32_t h5 = lbase + ((ax1 ^ ay0 ^ az1) & NGP_MASK);
    const uint32_t h6 = lbase + ((ax0 ^ ay1 ^ az1) & NGP_MASK);
    const uint32_t h7 = lbase + ((ax1 ^ ay1 ^ az1) & NGP_MASK);

    // issue all 8 gathers (global_load_b64) before consuming
    const v2f v0 = tab[h0];
    const v2f v1 = tab[h1];
    const v2f v2 = tab[h2];
    const v2f v3 = tab[h3];
    const v2f v4 = tab[h4];
    const v2f v5 = tab[h5];
    const v2f v6 = tab[h6];
    const v2f v7 = tab[h7];

    const float wx0 = 1.0f - wx, wy0 = 1.0f - wy, wz0 = 1.0f - wz;

    v2f acc = (wx0 * wy0 * wz0) * v0;
    acc += (wx  * wy0 * wz0) * v1;
    acc += (wx0 * wy  * wz0) * v2;
    acc += (wx  * wy  * wz0) * v3;
    acc += (wx0 * wy0 * wz ) * v4;
    acc += (wx  * wy0 * wz ) * v5;
    acc += (wx0 * wy  * wz ) * v6;
    acc += (wx  * wy  * wz ) * v7;

    // out row = 32 floats = 16 x float2; NT store: streaming, keep L2 for tables
    v2f* __restrict__ outv = reinterpret_cast<v2f*>(out);
    __builtin_nontemporal_store(acc,
        &outv[(size_t)(basePt + ptLocal) * NGP_N_LEVELS + lvl]);
}

extern "C" void kernel_launch(void* const* d_in, const int* in_sizes, int n_in,
                              void* d_out, int out_size, void* d_ws, size_t ws_size,
                              hipStream_t stream) {
    (void)n_in; (void)out_size; (void)d_ws; (void)ws_size;
    const float* x   = (const float*)d_in[0];   // (B,3) f32
    const float* emb = (const float*)d_in[1];   // (16, 2^19, 2) f32
    float* out       = (float*)d_out;           // (B,32) f32

    const int npts   = in_sizes[0] / 3;
    const int blocks = (npts + NGP_PTS_PER_BLK - 1) / NGP_PTS_PER_BLK;

    hipLaunchKernelGGL(HashEmbedding_ngp_kernel,
                       dim3(blocks), dim3(NGP_THREADS), 0, stream,
                       x, emb, out, npts);
}